// NeighbourChannels_14516989460712
// MI455X (gfx1250) — compile-verified
//
#include <hip/hip_runtime.h>
#include <stdint.h>

// ---- vector types ----
typedef float        f32x2 __attribute__((ext_vector_type(2)));
typedef float        f32x4 __attribute__((ext_vector_type(4)));
typedef float        f32x8 __attribute__((ext_vector_type(8)));
typedef unsigned int u32x4 __attribute__((ext_vector_type(4)));
typedef int          i32x4 __attribute__((ext_vector_type(4)));
typedef int          i32x8 __attribute__((ext_vector_type(8)));

#define CH      256      // channels
#define TS      128      // spatial tile (columns) per workgroup
#define STRIDE  136      // LDS row stride in floats (128 data + 8 pad by TDM)
#define HW      16384    // 128*128 spatial positions per plane
#define NBATCH  16

// out[b,c,s] = colsum[b,s] - x[b,c,s]
__global__ __launch_bounds__(256) void neighbour_channels_kernel(
    const float* __restrict__ x, float* __restrict__ out) {
  __shared__ __align__(16) float tile[CH * STRIDE];  // 139,264 B
  __shared__ __align__(16) float csum[TS];

  const int tid  = threadIdx.x;
  const int lane = tid & 31;
  const int wave = __builtin_amdgcn_readfirstlane(tid >> 5);  // 0..7, SGPR

  const int b  = blockIdx.x >> 7;             // 128 spatial blocks per batch
  const int s0 = (blockIdx.x & 127) * TS;     // spatial start of this tile
  const size_t plane = (size_t)b * CH * HW + (size_t)s0;

  // ---------- Phase 1: TDM DMA global -> LDS ----------
  // Each wave issues one tensor_load_to_lds for a 32(ch) x 128(spatial) slab.
  // Tile rows are 128 floats (512 B); TDM pads 8 dwords after every 128 dwords
  // so LDS rows land at stride 136 floats (bank-decorrelated half-waves).
  {
    const uint64_t ga = (uint64_t)(uintptr_t)(x + plane + (size_t)wave * 32 * HW);
    const uint32_t la = (uint32_t)(uintptr_t)(&tile[0]) +
                        (uint32_t)wave * 32u * STRIDE * 4u;

    u32x4 g0;
    g0[0] = 1u;                                           // count=1, user mode
    g0[1] = la;                                           // lds_addr (bytes)
    g0[2] = (uint32_t)ga;                                 // global_addr[31:0]
    g0[3] = (uint32_t)((ga >> 32) & 0x1FFFFFFu)           // global_addr[56:32]
          | (2u << 30);                                   // type = 2 ("image")

    i32x8 g1;
    g1[0] = (2 << 16)       // data_size = 4 bytes
          | (1 << 20)       // pad_enable
          | (6 << 22)       // pad_interval = 128 dwords (one row)
          | (7 << 25);      // pad_amount   = 8 dwords
    g1[1] = (HW & 0xFFFF) << 16;                   // tensor_dim0[15:0]
    g1[2] = ((HW >> 16) & 0xFFFF) | (32 << 16);    // tensor_dim0[31:16] | tensor_dim1 lo
    g1[3] = 0 | (128 << 16);                       // tensor_dim1 hi | tile_dim0 = 128
    g1[4] = 32;                                    // tile_dim1 = 32, tile_dim2 = 0
    g1[5] = HW;                                    // tensor_dim0_stride[31:0]
    g1[6] = 0;                                     // dim0_stride hi | dim1_stride lo
    g1[7] = 0;                                     // tensor_dim1_stride hi

    i32x4 g2;
    g2[0] = 1;              // tensor_dim2 = 1 (degenerate)
    g2[1] = 0;              // tensor_dim3
    g2[2] = 0;              // tensor_dim2_stride lo
    g2[3] = 0;              // stride hi | tile_dim3 = 0
    i32x4 g3 = {0, 0, 0, 0};
    i32x8 g4 = {0, 0, 0, 0, 0, 0, 0, 0};   // extra group (clang-23 6-arg form): unused

    __builtin_amdgcn_tensor_load_to_lds(g0, g1, g2, g3, g4, 0);
    __builtin_amdgcn_s_wait_tensorcnt(0);
  }
  __syncthreads();

  // ---------- Phase 2: WMMA channel reduction ----------
  // A = ones(16x4)  =>  D[m,n] = sum_k B[k,n] (+C). Accumulate over 64 K-steps
  // of 4 channels each => colsum of 16 spatial columns per wave, in f32.
  // Invariant to B's K<->lane layout details since A is all-ones.
  {
    const int col   = (wave << 4) + (lane & 15);   // this lane's spatial column
    const int coff  = (lane >> 4) << 1;            // low half-wave: ch+0/1, high: ch+2/3
    f32x2 a; a[0] = 1.0f; a[1] = 1.0f;
    f32x8 acc = {0.f, 0.f, 0.f, 0.f, 0.f, 0.f, 0.f, 0.f};
#pragma unroll 8
    for (int k = 0; k < 64; ++k) {
      const int c0 = (k << 2) + coff;
      f32x2 bv;
      bv[0] = tile[c0 * STRIDE + col];
      bv[1] = tile[(c0 + 1) * STRIDE + col];
      acc = __builtin_amdgcn_wmma_f32_16x16x4_f32(
          /*neg_a=*/false, a, /*neg_b=*/false, bv,
          /*c_mod=*/(short)0, acc, /*reuse_a=*/false, /*reuse_b=*/false);
    }
    if (lane < 16) csum[(wave << 4) + lane] = acc[0];  // D vgpr0, lanes 0..15 = colsum[n]
  }
  __syncthreads();

  // ---------- Phase 3: out = colsum - x, streamed with NT b128 stores ----------
  // 256 threads x 32 iters x float4 = 32768 floats. Per-iter: 32 consecutive
  // threads cover one 512 B contiguous row segment -> fully coalesced.
#pragma unroll 4
  for (int i = 0; i < 32; ++i) {
    const int idx = (i << 8) + tid;        // 0..8191 float4 tiles
    const int c   = idx >> 5;              // channel row (32 float4 per row)
    const int s4  = (idx & 31) << 2;       // float offset within row
    const f32x4 xv = *(const f32x4*)(&tile[c * STRIDE + s4]);
    const f32x4 cv = *(const f32x4*)(&csum[s4]);
    const f32x4 ov = cv - xv;
    __builtin_nontemporal_store(ov, (f32x4*)(out + plane + (size_t)c * HW + s4));
  }
}

extern "C" void kernel_launch(void* const* d_in, const int* in_sizes, int n_in,
                              void* d_out, int out_size, void* d_ws, size_t ws_size,
                              hipStream_t stream) {
  (void)in_sizes; (void)n_in; (void)out_size; (void)d_ws; (void)ws_size;
  const float* x = (const float*)d_in[0];
  float* out     = (float*)d_out;
  dim3 grid(NBATCH * (HW / TS));   // 16 * 128 = 2048 workgroups
  neighbour_channels_kernel<<<grid, 256, 0, stream>>>(x, out);
}